// GNNEncoder_43413529428079
// MI455X (gfx1250) — compile-verified
//
#include <hip/hip_runtime.h>
#include <math.h>

#define C 128          // feature width (fixed by reference)
#define TPB 256        // 8 waves per block (wave32)

typedef float v2f __attribute__((ext_vector_type(2)));
typedef float v8f __attribute__((ext_vector_type(8)));

// ------------------------------------------------------------------
// degree: deg[dst[e]] += 1  (self-loop +1 folded into dinv kernel)
// ------------------------------------------------------------------
__global__ void deg_kernel(const int* __restrict__ dst, float* __restrict__ deg, int E) {
    int e = blockIdx.x * blockDim.x + threadIdx.x;
    if (e < E) atomicAdd(&deg[dst[e]], 1.0f);
}

__global__ void dinv_kernel(float* __restrict__ degdinv, int N) {
    int i = blockIdx.x * blockDim.x + threadIdx.x;
    if (i < N) degdinv[i] = rsqrtf(degdinv[i] + 1.0f);   // deg incl. self-loop, always > 0
}

// ------------------------------------------------------------------
// Y[N,128] = X[N,128] @ W[128,128]  via V_WMMA_F32_16X16X4_F32.
// One wave -> one 16-row strip x all 8 col tiles (8 v8f accumulators).
// W staged in LDS in 64-row K-chunks, "paired-K" layout so a B fragment
// is one 8-byte LDS load:  Wl[(k>>1)*256 + n*2 + (k&1)]
// ------------------------------------------------------------------
__global__ __launch_bounds__(TPB) void gemm_wmma(const float* __restrict__ X,
                                                 const float* __restrict__ W,
                                                 float* __restrict__ Y,
                                                 int nStrips) {
    __shared__ float Wl[64 * C];   // 32 KB chunk of W (64 K-rows)

    const int wave = threadIdx.x >> 5;
    const int lane = threadIdx.x & 31;
    const int half = lane >> 4;    // 0: lanes 0-15, 1: lanes 16-31
    const int lo   = lane & 15;

    const int strip  = blockIdx.x * (TPB / 32) + wave;
    const bool active = (strip < nStrips);
    const int m0 = strip * 16;

    v8f acc[8] = {};               // 8 column tiles of 16x16 f32

    const float* Arow = X + (size_t)(active ? (m0 + lo) : 0) * C + 2 * half;

    for (int kc = 0; kc < C; kc += 64) {
        __syncthreads();           // protect previous chunk's readers
        // cooperative load of W rows [kc, kc+64) into paired-K layout
        for (int idx = threadIdx.x; idx < 64 * C; idx += TPB) {
            int k = idx >> 7;              // 0..63 (local K row)
            int n = idx & (C - 1);
            Wl[(k >> 1) * 256 + n * 2 + (k & 1)] = W[(size_t)(kc + k) * C + n];
        }
        __syncthreads();

        if (active) {
            #pragma unroll
            for (int k0 = 0; k0 < 64; k0 += 4) {
                // A fragment: row m0+lo, K = kc+k0 + {0,1} (lanes 0-15) / {2,3} (lanes 16-31)
                v2f a = *(const v2f*)(Arow + kc + k0);
                const float* Bbase = &Wl[((k0 >> 1) + half) * 256 + lo * 2];
                #pragma unroll
                for (int t = 0; t < 8; ++t) {
                    v2f b = *(const v2f*)(Bbase + t * 32);   // cols t*16+lo, K pair
                    acc[t] = __builtin_amdgcn_wmma_f32_16x16x4_f32(
                        false, a, false, b, (short)0, acc[t], false, false);
                }
            }
        }
    }

    if (active) {
        float* Yt = Y + (size_t)m0 * C;
        #pragma unroll
        for (int t = 0; t < 8; ++t)
            #pragma unroll
            for (int r = 0; r < 8; ++r)     // VGPR r -> rows r (lanes 0-15), r+8 (lanes 16-31)
                Yt[(size_t)(r + 8 * half) * C + t * 16 + lo] = acc[t][r];
    }
}

// scalar tail for row counts not divisible by 16 (unused when N%16==0)
__global__ void gemm_tail(const float* __restrict__ X, const float* __restrict__ W,
                          float* __restrict__ Y, int row0, int rows) {
    int i = blockIdx.x * blockDim.x + threadIdx.x;
    if (i >= rows * C) return;
    int r = row0 + i / C, n = i & (C - 1);
    float s = 0.f;
    for (int k = 0; k < C; ++k) s += X[(size_t)r * C + k] * W[(size_t)k * C + n];
    Y[(size_t)r * C + n] = s;
}

// ------------------------------------------------------------------
// out[dst] += h[src] * (dinv[src]*dinv[dst])  — one wave per edge,
// lane handles 4 channels: float4 gather + 4x global_atomic_add_f32
// (destination buffer is L2-resident: 25.6 MB << 192 MB L2)
// edges [0,E) from edge_index, [E, E+N) are self-loops.
// ------------------------------------------------------------------
__global__ void agg_kernel(const float* __restrict__ H, const int* __restrict__ src,
                           const int* __restrict__ dst, const float* __restrict__ dinv,
                           float* __restrict__ out, int E, int N) {
    int gw   = (blockIdx.x * blockDim.x + threadIdx.x) >> 5;
    int lane = threadIdx.x & 31;
    int nw   = (gridDim.x * blockDim.x) >> 5;
    int total = E + N;
    for (int e = gw; e < total; e += nw) {
        int s, d;
        if (e < E) { s = src[e]; d = dst[e]; } else { s = e - E; d = s; }
        float nrm = dinv[s] * dinv[d];
        float4 v = *(const float4*)(H + (size_t)s * C + lane * 4);
        float* o = out + (size_t)d * C + lane * 4;
        atomicAdd(o + 0, v.x * nrm);
        atomicAdd(o + 1, v.y * nrm);
        atomicAdd(o + 2, v.z * nrm);
        atomicAdd(o + 3, v.w * nrm);
    }
}

// ------------------------------------------------------------------
// out = ELU(in + b)   (alpha = 1), float4 vectorized
// ------------------------------------------------------------------
__global__ void bias_elu_kernel(const float* __restrict__ in, const float* __restrict__ b,
                                float* __restrict__ out, size_t n4) {
    size_t i = (size_t)blockIdx.x * blockDim.x + threadIdx.x;
    if (i >= n4) return;
    size_t idx = i * 4;
    int c = (int)(idx & (C - 1));
    float4 v = *(const float4*)(in + idx);
    v.x += b[c + 0]; v.y += b[c + 1]; v.z += b[c + 2]; v.w += b[c + 3];
    v.x = v.x > 0.f ? v.x : (expf(v.x) - 1.f);
    v.y = v.y > 0.f ? v.y : (expf(v.y) - 1.f);
    v.z = v.z > 0.f ? v.z : (expf(v.z) - 1.f);
    v.w = v.w > 0.f ? v.w : (expf(v.w) - 1.f);
    *(float4*)(out + idx) = v;
}

// ------------------------------------------------------------------
extern "C" void kernel_launch(void* const* d_in, const int* in_sizes, int n_in,
                              void* d_out, int out_size, void* d_ws, size_t ws_size,
                              hipStream_t stream) {
    const float* x  = (const float*)d_in[0];
    const int*   ei = (const int*)d_in[1];     // [2, E] flat: row0 = src, row1 = dst
    const float* W1 = (const float*)d_in[2];
    const float* b1 = (const float*)d_in[3];
    const float* W2 = (const float*)d_in[4];
    const float* b2 = (const float*)d_in[5];

    const int N = in_sizes[0] / C;
    const int E = in_sizes[1] / 2;
    const int* srcI = ei;
    const int* dstI = ei + E;

    // workspace layout: dinv[N] | buf0[N*C] | buf1[N*C]
    char* ws = (char*)d_ws;
    float* dinv = (float*)ws;
    size_t o1 = ((size_t)N * 4 + 255) & ~(size_t)255;
    size_t hbytes = ((size_t)N * C * 4 + 255) & ~(size_t)255;
    float* buf0 = (float*)(ws + o1);
    float* buf1 = (float*)(ws + o1 + hbytes);

    const int strips   = N / 16;
    const int tailRow0 = strips * 16;
    const int tailRows = N - tailRow0;
    const dim3 blk(TPB);
    const int gemmBlocks = (strips + (TPB / 32) - 1) / (TPB / 32);
    const int aggBlocks  = 4096;                       // 32768 waves, grid-stride over E+N
    const size_t n4      = (size_t)N * C / 4;
    const int eluBlocks  = (int)((n4 + TPB - 1) / TPB);

    // ---- normalization coefficients ----
    hipMemsetAsync(dinv, 0, (size_t)N * 4, stream);
    deg_kernel<<<(E + TPB - 1) / TPB, blk, 0, stream>>>(dstI, dinv, E);
    dinv_kernel<<<(N + TPB - 1) / TPB, blk, 0, stream>>>(dinv, N);

    // ---- layer 1 ----
    gemm_wmma<<<gemmBlocks, blk, 0, stream>>>(x, W1, buf0, strips);
    if (tailRows > 0)
        gemm_tail<<<(tailRows * C + TPB - 1) / TPB, blk, 0, stream>>>(x, W1, buf0, tailRow0, tailRows);
    hipMemsetAsync(buf1, 0, (size_t)N * C * 4, stream);
    agg_kernel<<<aggBlocks, blk, 0, stream>>>(buf0, srcI, dstI, dinv, buf1, E, N);
    bias_elu_kernel<<<eluBlocks, blk, 0, stream>>>(buf1, b1, buf0, n4);

    // ---- layer 2 ----
    gemm_wmma<<<gemmBlocks, blk, 0, stream>>>(buf0, W2, buf1, strips);
    if (tailRows > 0)
        gemm_tail<<<(tailRows * C + TPB - 1) / TPB, blk, 0, stream>>>(buf0, W2, buf1, tailRow0, tailRows);
    hipMemsetAsync(buf0, 0, (size_t)N * C * 4, stream);
    agg_kernel<<<aggBlocks, blk, 0, stream>>>(buf1, srcI, dstI, dinv, buf0, E, N);
    bias_elu_kernel<<<eluBlocks, blk, 0, stream>>>(buf0, b2, (float*)d_out, n4);
}